// GCNCoxModel_1786706395457
// MI455X (gfx1250) — compile-verified
//
#include <hip/hip_runtime.h>
#include <cstdint>

#define NFEAT 128

typedef __attribute__((ext_vector_type(2))) float v2f;
typedef __attribute__((ext_vector_type(8))) float v8f;

// ---------------- degree / norm kernels ----------------

__global__ void init_deg_kernel(float* __restrict__ deg, int n) {
    int i = blockIdx.x * blockDim.x + threadIdx.x;
    if (i < n) deg[i] = 1.0f;   // self-loop contributes 1 to every node's degree
}

__global__ void deg_accum_kernel(const int* __restrict__ dst, float* __restrict__ deg, int e) {
    int i = blockIdx.x * blockDim.x + threadIdx.x;
    if (i < e) atomicAdd(&deg[dst[i]], 1.0f);
}

__global__ void dinv_kernel(float* __restrict__ deg, int n) {
    int i = blockIdx.x * blockDim.x + threadIdx.x;
    if (i < n) deg[i] = rsqrtf(deg[i]);   // deg >= 1 always (self-loop)
}

// ---------------- xw = x @ W^T via V_WMMA_F32_16X16X4_F32 ----------------
// Block = 256 threads = 8 waves. Block handles a 16-row node tile (in LDS,
// shared A operand); wave w handles output columns [16w, 16w+16).
// Requires n % 16 == 0 (true for N=100000).

__global__ __launch_bounds__(256) void gemm_xwT_wmma(const float* __restrict__ x,
                                                     const float* __restrict__ W,
                                                     float* __restrict__ xw, int n) {
    __shared__ float lds_x[16 * NFEAT];   // 8 KB: 16 node rows x 128 feats

    const int tid  = threadIdx.x;
    const int n0   = blockIdx.x * 16;

    // Cooperative coalesced load of the 16x128 A tile into LDS (512 float4s).
    const float4* xs = (const float4*)(x + (size_t)n0 * NFEAT);
    float4* ls = (float4*)lds_x;
    ls[tid]       = xs[tid];
    ls[tid + 256] = xs[tid + 256];
    __syncthreads();

    const int wave  = tid >> 5;
    const int lane  = tid & 31;
    const int o0    = wave * 16;         // this wave's output-column tile
    const int row   = lane & 15;         // M index (A) / N index (B)
    const int khalf = (lane >> 4) << 1;  // lanes 0-15 -> K offset 0; lanes 16-31 -> 2

    v8f acc = {};
    const float* wrow = W + (size_t)(o0 + row) * NFEAT;  // B[K][N] = W[o0+N][K]

#pragma unroll
    for (int k0 = 0; k0 < NFEAT; k0 += 4) {
        const int kk = k0 + khalf;
        v2f a, b;
        a.x = lds_x[row * NFEAT + kk];       // A vgpr0: K=k0 / k0+2
        a.y = lds_x[row * NFEAT + kk + 1];   // A vgpr1: K=k0+1 / k0+3
        b.x = wrow[kk];                      // B vgpr0: K=k0 / k0+2
        b.y = wrow[kk + 1];                  // B vgpr1: K=k0+1 / k0+3
        // 8 args: (neg_a, A, neg_b, B, c_mod, C, reuse_a, reuse_b)
        acc = __builtin_amdgcn_wmma_f32_16x16x4_f32(false, a, false, b,
                                                    (short)0, acc, false, false);
    }

    // D layout: VGPR r -> M = r + 8*(lane/16), N = lane & 15
    const int col = o0 + row;
#pragma unroll
    for (int r = 0; r < 8; ++r) {
        const int m = r + ((lane >> 4) << 3);
        xw[(size_t)(n0 + m) * NFEAT + col] = acc[r];
    }
}

// ---------------- agg init: bias + self-loop term ----------------

__global__ void init_agg_kernel(const float* __restrict__ xw, const float* __restrict__ dinv,
                                const float* __restrict__ bias, float* __restrict__ agg,
                                int n) {
    size_t i = (size_t)blockIdx.x * blockDim.x + threadIdx.x;
    if (i < (size_t)n * NFEAT) {
        int node = (int)(i >> 7);
        int f    = (int)(i & (NFEAT - 1));
        float di = dinv[node];
        agg[i] = bias[f] + xw[i] * di * di;   // self-loop norm = dinv^2 = 1/deg
    }
}

// ---------------- edge scatter: agg[dst] += xw[src] * norm ----------------
// One wave32 per edge; each lane handles 4 contiguous features.

__global__ __launch_bounds__(256) void scatter_edges_kernel(const int* __restrict__ src,
                                                            const int* __restrict__ dst,
                                                            const float* __restrict__ xw,
                                                            const float* __restrict__ dinv,
                                                            float* __restrict__ agg, int e) {
    int gid  = blockIdx.x * blockDim.x + threadIdx.x;
    int edge = gid >> 5;
    int lane = gid & 31;
    if (edge >= e) return;

    int s = src[edge];
    int d = dst[edge];
    float w = dinv[s] * dinv[d];

    const float4* xs = (const float4*)(xw + (size_t)s * NFEAT);
    float4 v = xs[lane];
    float* ab = agg + (size_t)d * NFEAT + lane * 4;
    atomicAdd(ab + 0, v.x * w);
    atomicAdd(ab + 1, v.y * w);
    atomicAdd(ab + 2, v.z * w);
    atomicAdd(ab + 3, v.w * w);
}

// ---------------- finalize: relu + regression dot ----------------
// One wave32 per node; shuffle reduction across 32 lanes.

__global__ __launch_bounds__(256) void finalize_kernel(const float* __restrict__ agg,
                                                       const float* __restrict__ wreg,
                                                       const float* __restrict__ breg,
                                                       float* __restrict__ out, int n) {
    int gid  = blockIdx.x * blockDim.x + threadIdx.x;
    int node = gid >> 5;
    int lane = gid & 31;
    if (node >= n) return;

    const float4* av = (const float4*)(agg + (size_t)node * NFEAT);
    const float4* wv = (const float4*)wreg;
    float4 a = av[lane];
    float4 w = wv[lane];
    float dot = fmaxf(a.x, 0.0f) * w.x + fmaxf(a.y, 0.0f) * w.y +
                fmaxf(a.z, 0.0f) * w.z + fmaxf(a.w, 0.0f) * w.w;
#pragma unroll
    for (int off = 16; off > 0; off >>= 1)
        dot += __shfl_xor(dot, off, 32);
    if (lane == 0) out[node] = dot + breg[0];
}

// ---------------- launch ----------------

extern "C" void kernel_launch(void* const* d_in, const int* in_sizes, int n_in,
                              void* d_out, int out_size, void* d_ws, size_t ws_size,
                              hipStream_t stream) {
    const float* x    = (const float*)d_in[0];
    const int*   ei   = (const int*)d_in[1];   // [2, E]: row 0 = src, row 1 = dst
    const float* W    = (const float*)d_in[2];
    const float* b    = (const float*)d_in[3];
    const float* wreg = (const float*)d_in[4];
    const float* breg = (const float*)d_in[5];
    float* out = (float*)d_out;

    const int n = in_sizes[0] / NFEAT;
    const int e = in_sizes[1] / 2;
    const int* src = ei;
    const int* dst = ei + e;

    // workspace layout (all 16B-aligned for N=100000):
    float* deg = (float*)d_ws;                 // n floats; becomes dinv in-place
    float* xw  = deg + n;                      // n*128 floats
    float* agg = xw + (size_t)n * NFEAT;       // n*128 floats

    const int B = 256;
    init_deg_kernel<<<(n + B - 1) / B, B, 0, stream>>>(deg, n);
    deg_accum_kernel<<<(e + B - 1) / B, B, 0, stream>>>(dst, deg, e);
    dinv_kernel<<<(n + B - 1) / B, B, 0, stream>>>(deg, n);

    gemm_xwT_wmma<<<n / 16, B, 0, stream>>>(x, W, xw, n);

    unsigned int agg_blocks = (unsigned int)(((size_t)n * NFEAT + B - 1) / B);
    init_agg_kernel<<<agg_blocks, B, 0, stream>>>(xw, deg, b, agg, n);

    unsigned int sc_blocks = (unsigned int)(((size_t)e * 32 + B - 1) / B);
    scatter_edges_kernel<<<sc_blocks, B, 0, stream>>>(src, dst, xw, deg, agg, e);

    unsigned int fin_blocks = (unsigned int)(((size_t)n * 32 + B - 1) / B);
    finalize_kernel<<<fin_blocks, B, 0, stream>>>(agg, wreg, breg, out, n);
}